// WaveletTransform_55697135894581
// MI455X (gfx1250) — compile-verified
//
#include <hip/hip_runtime.h>
#include <stdint.h>

#define N_COLS     16384
#define FLEN       8
#define OUT_HALF   8199            // (16384+7+7)/2
#define ROW_OUT    (2*OUT_HALF)    // 16398
#define OPB        2048            // outputs (per filter) per block
#define OPT        8               // outputs per thread
#define NTHREADS   256
#define TILE       (2*OPB + 8)     // 4104 floats: covers x[2*M0-8 .. 2*M0+4095]
#define TILE4      (TILE/4)        // 1026 float4 slots

__device__ __forceinline__ void async_copy_f4(unsigned lds_off, const float4* gptr) {
#if defined(__HIP_DEVICE_COMPILE__)
    asm volatile("global_load_async_to_lds_b128 %0, %1, off"
                 :: "v"(lds_off), "v"(gptr) : "memory");
#endif
}

__device__ __forceinline__ void wait_asynccnt0() {
#if defined(__HIP_DEVICE_COMPILE__)
    asm volatile("s_wait_asynccnt 0" ::: "memory");
#endif
}

__global__ __launch_bounds__(NTHREADS)
void dwt_db4_rows_kernel(const float* __restrict__ x, float* __restrict__ out, int rows) {
    // db4 decomposition filters (pywt 'db4')
    const float LO[FLEN] = {-0.010597401784997278f,  0.032883011666982945f,
                             0.030841381835986965f, -0.18703481171888114f,
                            -0.02798376941698385f,   0.6308807679295904f,
                             0.7148465705525415f,    0.23037781330885523f};
    const float HI[FLEN] = {-0.23037781330885523f,   0.7148465705525415f,
                            -0.6308807679295904f,   -0.02798376941698385f,
                             0.18703481171888114f,   0.030841381835986965f,
                            -0.032883011666982945f, -0.010597401784997278f};

    __shared__ __align__(16) float tile[TILE];   // tile[p] = x_row[in0 + p] (zero outside row)

    const int row = blockIdx.y;
    const int M0  = blockIdx.x * OPB;            // first output index of this block
    const int in0 = 2 * M0 - 8;                  // first staged input index (8-float left pad)
    const int tid = threadIdx.x;

    const float* __restrict__ xrow = x + (size_t)row * N_COLS;

    // Clamped copy range; both ends are multiples of 8 floats -> b128-aligned everywhere.
    const int gstart = (in0 > 0) ? in0 : 0;
    const int gend   = (in0 + TILE < N_COLS) ? (in0 + TILE) : N_COLS;
    const int c0 = gstart - in0;                 // floats of left zero-pad  (0 or 8)
    const int c1 = gend   - in0;                 // end of valid data in tile

    const unsigned lds_base = (unsigned)(unsigned long long)(uintptr_t)&tile[0];
    float4* t4 = reinterpret_cast<float4*>(tile);

    // Zero-fill pad regions (disjoint from async-copy target; no-op for interior tiles).
    const float4 z = make_float4(0.f, 0.f, 0.f, 0.f);
    for (int i = tid; i < (c0 >> 2); i += NTHREADS) t4[i] = z;            // ds_store_b128
    for (int i = (c1 >> 2) + tid; i < TILE4; i += NTHREADS) t4[i] = z;    // ds_store_b128

    // CDNA5 async global->LDS DMA for the valid range (ASYNCcnt-tracked, 16B aligned).
    {
        const float4* g4 = reinterpret_cast<const float4*>(xrow + gstart);
        const unsigned lds0 = lds_base + (unsigned)(c0 * 4);
        const int n4 = (c1 - c0) >> 2;
        #pragma unroll 2
        for (int i = tid; i < n4; i += NTHREADS)
            async_copy_f4(lds0 + 16u * (unsigned)i, g4 + i);
        wait_asynccnt0();                        // each wave drains its own ASYNCcnt
    }
    __syncthreads();                             // all waves' zeros + transfers visible

    // ---- compute 8 outputs per thread from a 24-float register window ----
    const int m0 = M0 + tid * OPT;               // first output index of this thread
    float v[24];
    const float4* lw = reinterpret_cast<const float4*>(&tile[16 * tid]); // 64B-aligned
    #pragma unroll
    for (int j = 0; j < 6; ++j) {
        float4 q = lw[j];                        // ds_load_b128
        v[4*j + 0] = q.x; v[4*j + 1] = q.y; v[4*j + 2] = q.z; v[4*j + 3] = q.w;
    }
    // v[j] = x[2*m0 - 8 + j]  =>  x[2*(m0+mm)+1-k] = v[2*mm + 9 - k]
    float a[OPT], d[OPT];
    #pragma unroll
    for (int mm = 0; mm < OPT; ++mm) {
        float ca = 0.0f, cd = 0.0f;
        #pragma unroll
        for (int k = 0; k < FLEN; ++k) {
            float xv = v[2*mm + 9 - k];
            ca = fmaf(LO[k], xv, ca);
            cd = fmaf(HI[k], xv, cd);
        }
        a[mm] = ca; d[mm] = cd;
    }

    float* __restrict__ orow = out + (size_t)row * ROW_OUT;
    // cA: offset row*ROW_OUT + m0 is even -> 8B-aligned float2 stores
    #pragma unroll
    for (int p = 0; p < OPT / 2; ++p) {
        int m = m0 + 2 * p;
        if (m + 1 < OUT_HALF) {
            *reinterpret_cast<float2*>(orow + m) = make_float2(a[2*p], a[2*p + 1]);
        } else if (m < OUT_HALF) {
            orow[m] = a[2*p];
        }
    }
    // cD: +OUT_HALF (odd) breaks 8B alignment -> scalar stores
    #pragma unroll
    for (int mm = 0; mm < OPT; ++mm) {
        int m = m0 + mm;
        if (m < OUT_HALF) orow[OUT_HALF + m] = d[mm];
    }
}

extern "C" void kernel_launch(void* const* d_in, const int* in_sizes, int n_in,
                              void* d_out, int out_size, void* d_ws, size_t ws_size,
                              hipStream_t stream) {
    const float* x = (const float*)d_in[0];
    float* out = (float*)d_out;
    const int rows = in_sizes[0] / N_COLS;                 // 4096
    const int tiles = (OUT_HALF + OPB - 1) / OPB;          // 5
    dim3 grid(tiles, rows);
    dim3 block(NTHREADS);
    dwt_db4_rows_kernel<<<grid, block, 0, stream>>>(x, out, rows);
}